// TraceClassifier_22359599743197
// MI455X (gfx1250) — compile-verified
//
#include <hip/hip_runtime.h>
#include <hip/hip_bf16.h>
#include <math.h>

typedef __attribute__((ext_vector_type(2))) float v2f;
typedef __attribute__((ext_vector_type(8))) float v8f;

#define CDIV(a,b) (((a)+(b)-1)/(b))

// ---------------------------------------------------------------------------
// Generic fused GEMM: Out[M x Nout] = act( rowScale .* (preRelu(A[M x K]) @ W[Nout x K]^T)
//                                          + bias + Cadd )
// REQUIRES: M % 16 == 0, K % 4 == 0; Nout % (16*NT) == 0.
// One wave computes a 16 x (16*NT) tile via V_WMMA_F32_16X16X4_F32 (fp32 path):
// per K-step it loads the A pair ONCE and reuses it across NT WMMAs (NT=4 for all
// large GEMMs -> 5 loads per 4 WMMAs instead of 8).
// A-layout: lane&15 = M row; lane>>4 selects K pair {0,1}/{2,3} (ISA 7.12.2).
// B-layout: lane&15 = N col; same K split. C/D: element (m = v + 8*half, n = lane&15).
// rowScale is applied to the finished accumulator (distributes over the K sum).
// PRERELU is a template param so the common path has a pure load->wmma loop.
// ---------------------------------------------------------------------------
template <int NT, int PRERELU>
__global__ __launch_bounds__(256) void k_gemm(
    const float* __restrict__ A, const float* __restrict__ W,
    const float* __restrict__ bias, const float* __restrict__ Cadd,
    const float* __restrict__ rowScale, float* __restrict__ Out,
    int M, int K, int Nout, int act)
{
  int wavesPerBlock = blockDim.x >> 5;
  int wave = blockIdx.x * wavesPerBlock + ((int)threadIdx.x >> 5);
  int tilesNg = (Nout >> 4) / NT;           // groups of NT column-tiles
  int tiles   = (M >> 4) * tilesNg;
  if (wave >= tiles) return;                // wave-uniform: EXEC stays all-ones for WMMA
  int tM = wave / tilesNg, tNg = wave % tilesNg;
  int rb = tM << 4, cb = tNg * (NT << 4);
  int lane = (int)threadIdx.x & 31;
  int half = lane >> 4;
  int l15  = lane & 15;
  int am = rb + l15;                        // A row handled by this lane

  const float* Arow = A + (size_t)am * K + (half << 1);
  const float* Wrow[NT];
  #pragma unroll
  for (int j = 0; j < NT; ++j)
    Wrow[j] = W + (size_t)(cb + (j << 4) + l15) * K + (half << 1);

  v8f c[NT];
  #pragma unroll
  for (int j = 0; j < NT; ++j)
    #pragma unroll
    for (int v = 0; v < 8; ++v) c[j][v] = 0.0f;

  #pragma unroll 2
  for (int k0 = 0; k0 < K; k0 += 4) {
    float2 a2 = *reinterpret_cast<const float2*>(Arow + k0);
    float a0 = a2.x, a1 = a2.y;
    if (PRERELU) { a0 = fmaxf(a0, 0.0f); a1 = fmaxf(a1, 0.0f); }
    v2f av; av.x = a0; av.y = a1;
    float2 b2[NT];
    #pragma unroll
    for (int j = 0; j < NT; ++j)
      b2[j] = *reinterpret_cast<const float2*>(Wrow[j] + k0);
    #pragma unroll
    for (int j = 0; j < NT; ++j) {
      v2f bv; bv.x = b2[j].x; bv.y = b2[j].y;
      c[j] = __builtin_amdgcn_wmma_f32_16x16x4_f32(false, av, false, bv, (short)0,
                                                   c[j], false, false);
    }
  }

  #pragma unroll
  for (int j = 0; j < NT; ++j) {
    int bn = cb + (j << 4) + l15;
    float bi = bias ? bias[bn] : 0.0f;
    #pragma unroll
    for (int v = 0; v < 8; ++v) {
      int mc = rb + (half << 3) + v;
      float o = c[j][v];
      if (rowScale) o *= rowScale[mc];      // scale distributes over the K sum
      o += bi;
      if (Cadd) o += Cadd[(size_t)mc * Nout + bn];
      if (act == 1) o = fmaxf(o, 0.0f);
      else if (act == 2) o = 1.0f / (1.0f + expf(-o));
      Out[(size_t)mc * Nout + bn] = o;
    }
  }
}

// ---------------------------------------------------------------------------
__global__ void k_zero(float* p, long long n) {
  long long t = (long long)blockIdx.x * blockDim.x + threadIdx.x;
  if (t < n) p[t] = 0.0f;
}

// feats[N x 192] = [emb_api | emb_status | emb_node | emb_depth | emb_pos | lat_h]
// One wave per node; lane = channel 0..31. lat_h = relu(lat*w1+b1) @ w2^T + b2.
__global__ __launch_bounds__(256) void k_feats(
    const float* __restrict__ lat,
    const int* __restrict__ api, const int* __restrict__ status,
    const int* __restrict__ node, const int* __restrict__ depth,
    const int* __restrict__ pos,
    const float* __restrict__ e_api, const float* __restrict__ e_status,
    const float* __restrict__ e_node, const float* __restrict__ e_depth,
    const float* __restrict__ e_pos,
    const float* __restrict__ w1, const float* __restrict__ b1,
    const float* __restrict__ w2, const float* __restrict__ b2,
    float* __restrict__ feats, int N)
{
  int wid  = ((int)blockIdx.x * blockDim.x + (int)threadIdx.x) >> 5;
  int lane = (int)threadIdx.x & 31;
  if (wid >= N) return;
  float t = fmaxf(lat[wid] * w1[lane] + b1[lane], 0.0f);
  float acc = b2[lane];
  #pragma unroll
  for (int e = 0; e < 32; ++e) acc += __shfl(t, e, 32) * w2[lane * 32 + e];
  float* f = feats + (size_t)wid * 192;
  f[lane]       = e_api[(size_t)api[wid] * 32 + lane];
  f[32 + lane]  = e_status[(size_t)status[wid] * 32 + lane];
  f[64 + lane]  = e_node[(size_t)node[wid] * 32 + lane];
  int dc = depth[wid]; dc = dc < 0 ? 0 : (dc > 63 ? 63 : dc);
  f[96 + lane]  = e_depth[(size_t)dc * 32 + lane];
  int pc = pos[wid]; pc = pc < 0 ? 0 : (pc > 511 ? 511 : pc);
  f[128 + lane] = e_pos[(size_t)pc * 32 + lane];
  f[160 + lane] = acc;
}

// degree counting (float atomics; counts are small exact integers)
__global__ void k_count2(const int* __restrict__ s, const int* __restrict__ d,
                         float* __restrict__ deg, int E) {
  int t = (int)blockIdx.x * blockDim.x + (int)threadIdx.x;
  if (t >= E) return;
  atomicAdd(&deg[s[t]], 1.0f);
  atomicAdd(&deg[d[t]], 1.0f);
}
__global__ void k_count1(const int* __restrict__ s, float* __restrict__ deg, int E) {
  int t = (int)blockIdx.x * blockDim.x + (int)threadIdx.x;
  if (t >= E) return;
  atomicAdd(&deg[s[t]], 1.0f);
}
__global__ void k_finnorm(float* __restrict__ p, float selfAdd, int N) {
  int t = (int)blockIdx.x * blockDim.x + (int)threadIdx.x;
  if (t >= N) return;
  float d = fmaxf(p[t] + selfAdd, 1.0f);
  p[t] = 1.0f / sqrtf(d);
}

// y = x * norm[row]; optionally also initialize agg = y (covers self-loop edges)
__global__ void k_scale(const float* __restrict__ x, const float* __restrict__ norm,
                        float* __restrict__ y, float* __restrict__ agg, int N) {
  int t = (int)blockIdx.x * blockDim.x + (int)threadIdx.x;
  if (t >= N * 64) return;
  float v = x[t] * norm[t >> 6];
  y[t] = v;
  if (agg) agg[t] = v;
}

// symmetrized scatter: per listed edge add both directions (call graph)
__global__ void k_scatter_bi(const int* __restrict__ src, const int* __restrict__ dst,
                             const float* __restrict__ y, float* __restrict__ agg, int E) {
  int t = (int)blockIdx.x * blockDim.x + (int)threadIdx.x;
  if (t >= E * 16) return;
  int e = t >> 4, c = (t & 15) << 2;
  int s = src[e], d = dst[e];
  const float4 ys = *reinterpret_cast<const float4*>(y + (size_t)s * 64 + c);
  const float4 yd = *reinterpret_cast<const float4*>(y + (size_t)d * 64 + c);
  float* pd = agg + (size_t)d * 64 + c;
  float* ps = agg + (size_t)s * 64 + c;
  atomicAdd(pd + 0, ys.x); atomicAdd(pd + 1, ys.y);
  atomicAdd(pd + 2, ys.z); atomicAdd(pd + 3, ys.w);
  atomicAdd(ps + 0, yd.x); atomicAdd(ps + 1, yd.y);
  atomicAdd(ps + 2, yd.z); atomicAdd(ps + 3, yd.w);
}

// one-directional scatter over full edge list (host graph, loops included)
__global__ void k_scatter_uni(const int* __restrict__ src, const int* __restrict__ dst,
                              const float* __restrict__ y, float* __restrict__ agg, int E) {
  int t = (int)blockIdx.x * blockDim.x + (int)threadIdx.x;
  if (t >= E * 16) return;
  int e = t >> 4, c = (t & 15) << 2;
  int s = src[e], d = dst[e];
  const float4 ys = *reinterpret_cast<const float4*>(y + (size_t)s * 64 + c);
  float* pd = agg + (size_t)d * 64 + c;
  atomicAdd(pd + 0, ys.x); atomicAdd(pd + 1, ys.y);
  atomicAdd(pd + 2, ys.z); atomicAdd(pd + 3, ys.w);
}

// TreeLSTM parent gather: node p (local lp<64) sums its children (local lp*8+1 .. min(lp*8+8,511))
__global__ void k_msg(const float* __restrict__ h, const float* __restrict__ fe,
                      const float* __restrict__ cst,
                      float* __restrict__ msg_h, float* __restrict__ msg_fc, int N) {
  int t = (int)blockIdx.x * blockDim.x + (int)threadIdx.x;
  if (t >= N * 64) return;
  int node = t >> 6, ch = t & 63;
  int lp = node & 511;
  float sh = 0.0f, sfc = 0.0f;
  if (lp < 64) {
    int base = (node - lp) + lp * 8 + 1;   // first child (global index)
    int nc = 511 - lp * 8; if (nc > 8) nc = 8;
    for (int j = 0; j < nc; ++j) {
      size_t ci = (size_t)(base + j) * 64 + ch;
      sh  += h[ci];
      sfc += fe[ci] * cst[ci];
    }
  }
  msg_h[t] = sh;
  msg_fc[t] = sfc;
}

// TreeLSTM elementwise: cn = sig(i)*tanh(u)+msg_fc; hn = sig(o)*tanh(cn); commit at level lvl
__global__ void k_ew(const float* __restrict__ iou, const float* __restrict__ msg_fc,
                     const int* __restrict__ tlv, int lvl,
                     float* __restrict__ h, float* __restrict__ c, int N) {
  int t = (int)blockIdx.x * blockDim.x + (int)threadIdx.x;
  if (t >= N * 64) return;
  int node = t >> 6, ch = t & 63;
  size_t b = (size_t)node * 192;
  float iv = iou[b + ch], ov = iou[b + 64 + ch], uv = iou[b + 128 + ch];
  float si = 1.0f / (1.0f + expf(-iv));
  float so = 1.0f / (1.0f + expf(-ov));
  float cn = si * tanhf(uv) + msg_fc[t];
  float hn = so * tanhf(cn);
  if (tlv[node] == lvl) { h[t] = hn; c[t] = cn; }
}

// graph-mean pooling: P[g][0:192] = mean(h_call|h_host|h_tl); gctx[g][0:7] = mean(ctx)
__global__ __launch_bounds__(256) void k_pool(
    const float* __restrict__ h_call, const float* __restrict__ h_host,
    const float* __restrict__ h_tl, const float* __restrict__ ctx,
    float* __restrict__ P, float* __restrict__ gctx, int N) {
  int g = (int)blockIdx.x;
  int t = (int)threadIdx.x;
  size_t nb = (size_t)g * 512;
  if (t < 192) {
    const float* sp = (t < 64) ? (h_call + t)
                    : (t < 128) ? (h_host + (t - 64))
                                : (h_tl + (t - 128));
    float s = 0.0f;
    for (int n2 = 0; n2 < 512; ++n2) s += sp[(nb + n2) * 64];
    P[(size_t)g * 256 + t] = s * (1.0f / 512.0f);
  } else if (t < 199) {
    int k = t - 192;
    float s = 0.0f;
    for (int n2 = 0; n2 < 512; ++n2) s += ctx[(nb + n2) * 7 + k];
    gctx[(size_t)g * 7 + k] = s * (1.0f / 512.0f);
  }
}

// ctx head (K=7 too small for WMMA): P[g][192+f] = relu(gctx[g] @ ctx_w^T + ctx_b)
__global__ void k_ctx(const float* __restrict__ gctx, const float* __restrict__ w,
                      const float* __restrict__ b, float* __restrict__ P, int G) {
  int t = (int)blockIdx.x * blockDim.x + (int)threadIdx.x;
  if (t >= G * 64) return;
  int g = t >> 6, f = t & 63;
  float acc = b[f];
  for (int k = 0; k < 7; ++k) acc += gctx[(size_t)g * 7 + k] * w[f * 7 + k];
  P[(size_t)g * 256 + 192 + f] = fmaxf(acc, 0.0f);
}

// pack hb_w/hc3_w/ht_w into one [16 x 128] head weight + [16] bias
__global__ void k_packhead(const float* __restrict__ hbw, const float* __restrict__ hbb,
                           const float* __restrict__ hc3w, const float* __restrict__ hc3b,
                           const float* __restrict__ htw, const float* __restrict__ htb,
                           float* __restrict__ headW, float* __restrict__ headB) {
  int t = (int)blockIdx.x * blockDim.x + (int)threadIdx.x;
  if (t >= 16 * 128) return;
  int r = t >> 7, k = t & 127;
  float v = (r == 0) ? hbw[k] : (r < 4) ? hc3w[(r - 1) * 128 + k] : htw[(r - 4) * 128 + k];
  headW[t] = v;
  if (k == 0) headB[r] = (r == 0) ? hbb[0] : (r < 4) ? hc3b[r - 1] : htb[r - 4];
}

// ---------------------------------------------------------------------------
static inline void gemm(hipStream_t s, const float* A, const float* W, const float* bias,
                        const float* Cadd, const float* rs, float* Out,
                        int M, int K, int Nout, int preRelu, int act) {
  if ((Nout & 63) == 0) {
    int tiles = (M >> 4) * ((Nout >> 4) >> 2);
    if (preRelu)
      k_gemm<4,1><<<CDIV(tiles, 8), 256, 0, s>>>(A, W, bias, Cadd, rs, Out, M, K, Nout, act);
    else
      k_gemm<4,0><<<CDIV(tiles, 8), 256, 0, s>>>(A, W, bias, Cadd, rs, Out, M, K, Nout, act);
  } else {
    int tiles = (M >> 4) * (Nout >> 4);
    k_gemm<1,0><<<CDIV(tiles, 8), 256, 0, s>>>(A, W, bias, Cadd, rs, Out, M, K, Nout, act);
  }
}

extern "C" void kernel_launch(void* const* d_in, const int* in_sizes, int n_in,
                              void* d_out, int out_size, void* d_ws, size_t ws_size,
                              hipStream_t stream) {
  // params (setup_inputs dict order)
  const float* emb_api    = (const float*)d_in[0];
  const float* emb_status = (const float*)d_in[1];
  const float* emb_node   = (const float*)d_in[2];
  const float* emb_depth  = (const float*)d_in[3];
  const float* emb_pos    = (const float*)d_in[4];
  const float* lat_w1     = (const float*)d_in[5];
  const float* lat_b1     = (const float*)d_in[6];
  const float* lat_w2     = (const float*)d_in[7];
  const float* lat_b2     = (const float*)d_in[8];
  const float* merge_w    = (const float*)d_in[9];
  const float* merge_b    = (const float*)d_in[10];
  const float* gcn1_w     = (const float*)d_in[11];
  const float* gcn1_b     = (const float*)d_in[12];
  const float* gcn2_w     = (const float*)d_in[13];
  const float* gcn2_b     = (const float*)d_in[14];
  const float* host1_w    = (const float*)d_in[15];
  const float* host1_b    = (const float*)d_in[16];
  const float* host2_w    = (const float*)d_in[17];
  const float* host2_b    = (const float*)d_in[18];
  const float* W_iouf     = (const float*)d_in[19];
  const float* U_iou      = (const float*)d_in[20];
  const float* b_iou      = (const float*)d_in[21];
  const float* Uf_w       = (const float*)d_in[22];
  const float* Uf_b       = (const float*)d_in[23];
  const float* tlout_w    = (const float*)d_in[24];
  const float* tlout_b    = (const float*)d_in[25];
  const float* ctx_w      = (const float*)d_in[26];
  const float* ctx_b      = (const float*)d_in[27];
  const float* fuse_w     = (const float*)d_in[28];
  const float* fuse_b     = (const float*)d_in[29];
  const float* hb_w       = (const float*)d_in[30];
  const float* hb_b       = (const float*)d_in[31];
  const float* hc3_w      = (const float*)d_in[32];
  const float* hc3_b      = (const float*)d_in[33];
  const float* ht_w       = (const float*)d_in[34];
  const float* ht_b       = (const float*)d_in[35];
  // data
  const float* lat        = (const float*)d_in[36];
  const float* ctx        = (const float*)d_in[37];
  const int* api_id       = (const int*)d_in[38];
  const int* status_id    = (const int*)d_in[39];
  const int* node_id      = (const int*)d_in[40];
  const int* depth        = (const int*)d_in[41];
  const int* pos          = (const int*)d_in[42];
  const int* src          = (const int*)d_in[43];
  const int* dst          = (const int*)d_in[44];
  const int* tree_level   = (const int*)d_in[46];
  const int* host_src     = (const int*)d_in[47];
  const int* host_dst     = (const int*)d_in[48];

  int N  = in_sizes[38];
  int E  = in_sizes[43];
  int Eh = in_sizes[47];
  int G  = N / 512;

  // workspace layout (floats)
  float* ws = (float*)d_ws;
  size_t n = (size_t)N;
  float* feats  = ws;                          // n*192 (reused as iou_x)
  float* iou    = feats  + n * 192;            // n*192
  float* x      = iou    + n * 192;            // n*64
  float* bufA   = x      + n * 64;             // y / h
  float* bufB   = bufA   + n * 64;             // agg / c
  float* bufC   = bufB   + n * 64;             // t1 / fe
  float* h_call = bufC   + n * 64;
  float* h_host = h_call + n * 64;
  float* bufD   = h_host + n * 64;             // msg_h then h_tl
  float* bufE   = bufD   + n * 64;             // msg_fc
  float* norm_c = bufE   + n * 64;             // n
  float* norm_h = norm_c + n;                  // n
  float* P      = norm_h + n;                  // G*256
  float* fusedB = P      + (size_t)G * 256;    // G*128
  float* gctx   = fusedB + (size_t)G * 128;    // G*7
  float* headW  = gctx   + (size_t)G * 7;      // 16*128
  float* headB  = headW  + 16 * 128;           // 16

  // 1. node features (gathers + lat MLP), one wave per node
  k_feats<<<CDIV(N * 32, 256), 256, 0, stream>>>(
      lat, api_id, status_id, node_id, depth, pos,
      emb_api, emb_status, emb_node, emb_depth, emb_pos,
      lat_w1, lat_b1, lat_w2, lat_b2, feats, N);

  // 2. x = relu(feats @ merge_w^T + merge_b)   [N x 64]
  gemm(stream, feats, merge_w, merge_b, nullptr, nullptr, x, N, 192, 64, 0, 1);

  // 3. iou_x = x @ W_iouf^T (first 192 rows)   [N x 192] -> reuse feats buffer
  gemm(stream, x, W_iouf, nullptr, nullptr, nullptr, feats, N, 64, 192, 0, 0);

  // 4. symmetric degrees -> D^-1/2 (self-loop +1 for call; host list includes loops)
  k_zero<<<CDIV((long long)N, 256), 256, 0, stream>>>(norm_c, N);
  k_zero<<<CDIV((long long)N, 256), 256, 0, stream>>>(norm_h, N);
  k_count2<<<CDIV(E, 256), 256, 0, stream>>>(src, dst, norm_c, E);
  k_count1<<<CDIV(Eh, 256), 256, 0, stream>>>(host_src, norm_h, Eh);
  k_finnorm<<<CDIV(N, 256), 256, 0, stream>>>(norm_c, 1.0f, N);
  k_finnorm<<<CDIV(N, 256), 256, 0, stream>>>(norm_h, 0.0f, N);

  // 5. call-graph GCN (2 layers); agg initialized to y handles self-loops
  k_scale<<<CDIV(N * 64, 256), 256, 0, stream>>>(x, norm_c, bufA, bufB, N);
  k_scatter_bi<<<CDIV(E * 16, 256), 256, 0, stream>>>(src, dst, bufA, bufB, E);
  gemm(stream, bufB, gcn1_w, gcn1_b, nullptr, norm_c, bufC, N, 64, 64, 0, 1);
  k_scale<<<CDIV(N * 64, 256), 256, 0, stream>>>(bufC, norm_c, bufA, bufB, N);
  k_scatter_bi<<<CDIV(E * 16, 256), 256, 0, stream>>>(src, dst, bufA, bufB, E);
  gemm(stream, bufB, gcn2_w, gcn2_b, nullptr, norm_c, h_call, N, 64, 64, 0, 0);

  // 6. host-graph GCN (edge list already symmetrized + loops -> agg starts at 0)
  k_scale<<<CDIV(N * 64, 256), 256, 0, stream>>>(x, norm_h, bufA, nullptr, N);
  k_zero<<<CDIV(n * 64, 256), 256, 0, stream>>>(bufB, (long long)n * 64);
  k_scatter_uni<<<CDIV(Eh * 16, 256), 256, 0, stream>>>(host_src, host_dst, bufA, bufB, Eh);
  gemm(stream, bufB, host1_w, host1_b, nullptr, norm_h, bufC, N, 64, 64, 0, 1);
  k_scale<<<CDIV(N * 64, 256), 256, 0, stream>>>(bufC, norm_h, bufA, nullptr, N);
  k_zero<<<CDIV(n * 64, 256), 256, 0, stream>>>(bufB, (long long)n * 64);
  k_scatter_uni<<<CDIV(Eh * 16, 256), 256, 0, stream>>>(host_src, host_dst, bufA, bufB, Eh);
  gemm(stream, bufB, host2_w, host2_b, nullptr, norm_h, h_host, N, 64, 64, 0, 0);

  // 7. ChildSum TreeLSTM, level-scheduled (h=bufA, c=bufB, fe=bufC)
  k_zero<<<CDIV(n * 64, 256), 256, 0, stream>>>(bufA, (long long)n * 64);
  k_zero<<<CDIV(n * 64, 256), 256, 0, stream>>>(bufB, (long long)n * 64);
  for (int lvl = 0; lvl < 8; ++lvl) {
    gemm(stream, bufA, Uf_w, Uf_b, nullptr, nullptr, bufC, N, 64, 64, 0, 2);     // fe = sigmoid
    k_msg<<<CDIV(N * 64, 256), 256, 0, stream>>>(bufA, bufC, bufB, bufD, bufE, N);
    gemm(stream, bufD, U_iou, b_iou, feats /*iou_x*/, nullptr, iou, N, 64, 192, 0, 0);
    k_ew<<<CDIV(N * 64, 256), 256, 0, stream>>>(iou, bufE, tree_level, lvl, bufA, bufB, N);
  }
  // h_tl = relu(h) @ tlout_w^T + tlout_b  -> bufD
  gemm(stream, bufA, tlout_w, tlout_b, nullptr, nullptr, bufD, N, 64, 64, 1, 0);

  // 8. pooling + ctx head + fuse + output heads
  k_pool<<<G, 256, 0, stream>>>(h_call, h_host, bufD, ctx, P, gctx, N);
  k_ctx<<<CDIV(G * 64, 256), 256, 0, stream>>>(gctx, ctx_w, ctx_b, P, G);
  gemm(stream, P, fuse_w, fuse_b, nullptr, nullptr, fusedB, G, 256, 128, 0, 1);
  k_packhead<<<8, 256, 0, stream>>>(hb_w, hb_b, hc3_w, hc3_b, ht_w, ht_b, headW, headB);
  gemm(stream, fusedB, headW, headB, nullptr, nullptr, (float*)d_out, G, 128, 16, 0, 0);
}